// GCN_Mutagenicity_66116726554995
// MI455X (gfx1250) — compile-verified
//
#include <hip/hip_runtime.h>
#include <hip/hip_bf16.h>

typedef __attribute__((ext_vector_type(2))) float v2f;
typedef __attribute__((ext_vector_type(8))) float v8f;

#define BN_EPS 1e-5f

__device__ __forceinline__ void atomic_add_f32(float* p, float v) {
    __hip_atomic_fetch_add(p, v, __ATOMIC_RELAXED, __HIP_MEMORY_SCOPE_AGENT);
}

// ---------------- degree / dinv ----------------
__global__ void deg_init_kernel(float* deg, int n) {
    int i = blockIdx.x * blockDim.x + threadIdx.x;
    if (i < n) deg[i] = 1.0f;          // self-loop contributes 1
}

__global__ void deg_acc_kernel(const int* __restrict__ col, const float* __restrict__ w,
                               float* deg, int ne) {
    int e = blockIdx.x * blockDim.x + threadIdx.x;
    if (e < ne) atomic_add_f32(&deg[col[e]], w[e]);
}

__global__ void deg_inv_kernel(float* deg, int n) {
    int i = blockIdx.x * blockDim.x + threadIdx.x;
    if (i < n) deg[i] = rsqrtf(deg[i]);
}

// ---------------- dense GEMM via WMMA f32 16x16x4 ----------------
// Y[n x 64] = X[n x K] @ W[K x 64].  K is a compile-time constant (14 or 64),
// K4 = K rounded up to a multiple of 4.  256 threads = 8 waves; each wave
// computes one 16x16 tile; block covers 32 rows x 64 cols.  Both W and the
// 32-row A tile are staged in LDS (zero-padded in K), so the unrolled K-loop
// is pure ds_load + v_wmma with EXEC all-ones throughout.
template <int K, int K4>
__global__ __launch_bounds__(256) void gemm_wmma_kernel(const float* __restrict__ X,
                                                        const float* __restrict__ Wm,
                                                        float* __restrict__ Y, int n) {
    __shared__ float sW[K4 * 64];     // [K4][64], zero-padded rows K..K4-1
    __shared__ float sA[32 * K4];     // [32][K4], zero-padded cols K..K4-1
    const int tid = threadIdx.x;
    const int rowBlockBase = blockIdx.x * 32;

    // cooperative load of W (coalesced over columns)
#pragma unroll
    for (int i = tid; i < K4 * 64; i += 256) {
        int kk = i >> 6, nn = i & 63;
        sW[i] = (kk < K) ? Wm[kk * 64 + nn] : 0.0f;
    }
    // cooperative load of the 32-row A tile (coalesced within rows)
#pragma unroll
    for (int i = tid; i < 32 * K4; i += 256) {
        int r = i / K4, k = i % K4;   // K4 is a power-of-two constant; folds to shifts
        int gr = rowBlockBase + r;
        if (gr >= n) gr = n - 1;
        sA[i] = (k < K) ? X[(size_t)gr * K + k] : 0.0f;
    }
    __syncthreads();

    const int wave = tid >> 5;
    const int lane = tid & 31;
    const int rowTile = wave >> 2;          // 0..1
    const int colTile = wave & 3;           // 0..3
    const int colBase = colTile * 16;
    const int laneHi = lane >> 4;           // 0: K pair lo, 1: K pair hi
    const int laneLo = lane & 15;

    const float* aRow = sA + (rowTile * 16 + laneLo) * K4;
    const float* bCol = sW + colBase + laneLo;

    v8f c = {};
#pragma unroll
    for (int k0 = 0; k0 < K4; k0 += 4) {
        const int ka = k0 + laneHi * 2;
        float2 av = *(const float2*)(aRow + ka);   // aligned ds_load_b64
        v2f a; a.x = av.x; a.y = av.y;
        v2f b;
        b.x = bCol[(ka    ) * 64];
        b.y = bCol[(ka + 1) * 64];
        c = __builtin_amdgcn_wmma_f32_16x16x4_f32(false, a, false, b, (short)0, c,
                                                  false, false);
    }

    // C layout: VGPR r -> row outRow0 + r, col colBase + laneLo.
    // Tile rows are contiguous, so one bounds test covers all 8 stores:
    // fast path is a single base address + 8 immediate-offset stores.
    const int outRow0 = rowBlockBase + rowTile * 16 + laneHi * 8;
    float* yb = Y + (size_t)outRow0 * 64 + colBase + laneLo;
    if (outRow0 + 7 < n) {
#pragma unroll
        for (int r = 0; r < 8; ++r) yb[r * 64] = c[r];
    } else {
#pragma unroll
        for (int r = 0; r < 8; ++r)
            if (outRow0 + r < n) yb[r * 64] = c[r];
    }
}

// ---------------- self-loop init: acc = dinv^2 * xw + bias ----------------
__global__ void selfloop_init_kernel(const float* __restrict__ xw, const float* __restrict__ dinv,
                                     const float* __restrict__ bias, float* __restrict__ acc,
                                     int n) {
    int idx = blockIdx.x * blockDim.x + threadIdx.x;
    if (idx < n * 64) {
        int i = idx >> 6, f = idx & 63;
        float di = dinv[i];
        acc[idx] = di * di * xw[idx] + bias[f];
    }
}

// ---------------- edge scatter: acc[col] += norm * xw[row] ----------------
// One wave (32 lanes) per edge; each lane handles 2 features (float2).
__global__ void edge_scatter_kernel(const int* __restrict__ row, const int* __restrict__ col,
                                    const float* __restrict__ w, const float* __restrict__ dinv,
                                    const float* __restrict__ xw, float* __restrict__ acc,
                                    int ne) {
    int t = blockIdx.x * blockDim.x + threadIdx.x;
    int e = t >> 5;
    int lane = t & 31;
    if (e < ne) {
        int r = row[e], c = col[e];
        float nrm = dinv[r] * w[e] * dinv[c];
        float2 v = ((const float2*)(xw + (size_t)r * 64))[lane];
        float* dst = acc + (size_t)c * 64 + lane * 2;
        atomic_add_f32(dst,     nrm * v.x);
        atomic_add_f32(dst + 1, nrm * v.y);
    }
}

// ---------------- BN stats: per-feature sum / sumsq ----------------
__global__ void bn_stats_kernel(const float* __restrict__ acc, float* __restrict__ sums, int n) {
    __shared__ float s1[256];
    __shared__ float s2[256];
    const int f = threadIdx.x & 63;
    const int sub = threadIdx.x >> 6;   // 0..3
    float sum = 0.f, sq = 0.f;
    for (int i = blockIdx.x * 4 + sub; i < n; i += gridDim.x * 4) {
        float v = acc[(size_t)i * 64 + f];
        sum += v;
        sq += v * v;
    }
    s1[threadIdx.x] = sum;
    s2[threadIdx.x] = sq;
    __syncthreads();
    if (sub == 0) {
        sum = s1[f] + s1[f + 64] + s1[f + 128] + s1[f + 192];
        sq  = s2[f] + s2[f + 64] + s2[f + 128] + s2[f + 192];
        atomic_add_f32(&sums[f], sum);
        atomic_add_f32(&sums[64 + f], sq);
    }
}

// ---------------- BN apply + ReLU (may be in-place) ----------------
__global__ void bn_relu_kernel(const float* __restrict__ acc, const float* __restrict__ sums,
                               const float* __restrict__ g, const float* __restrict__ bt,
                               float* __restrict__ out, int n) {
    int idx = blockIdx.x * blockDim.x + threadIdx.x;
    if (idx < n * 64) {
        int f = idx & 63;
        float inv_n = 1.0f / (float)n;
        float mean = sums[f] * inv_n;
        float var = sums[64 + f] * inv_n - mean * mean;
        float y = g[f] * (acc[idx] - mean) * rsqrtf(var + BN_EPS) + bt[f];
        out[idx] = fmaxf(y, 0.0f);
    }
}

// ---------------- zero helper ----------------
__global__ void zero_kernel(float* p, int n) {
    int i = blockIdx.x * blockDim.x + threadIdx.x;
    if (i < n) p[i] = 0.0f;
}

// ---------------- pooling: graph_emb[batch[i]] += h[i] ----------------
__global__ void pool_kernel(const float* __restrict__ h, const int* __restrict__ batch,
                            float* __restrict__ gemb, int n) {
    int t = blockIdx.x * blockDim.x + threadIdx.x;
    int i = t >> 5;
    int lane = t & 31;
    if (i < n) {
        int gidx = batch[i];
        float2 v = ((const float2*)(h + (size_t)i * 64))[lane];
        float* dst = gemb + (size_t)gidx * 64 + lane * 2;
        atomic_add_f32(dst,     v.x);
        atomic_add_f32(dst + 1, v.y);
    }
}

// ---------------- final FC: out = gemb @ fcW + fcb ----------------
__global__ void fc_kernel(const float* __restrict__ gemb, const float* __restrict__ fcW,
                          const float* __restrict__ fcb, float* __restrict__ out, int ng) {
    int t = blockIdx.x * blockDim.x + threadIdx.x;
    if (t < ng * 2) {
        int g = t >> 1, c = t & 1;
        float s = fcb[c];
        const float* gr = gemb + (size_t)g * 64;
#pragma unroll
        for (int f = 0; f < 64; ++f) s += gr[f] * fcW[f * 2 + c];
        out[t] = s;
    }
}

extern "C" void kernel_launch(void* const* d_in, const int* in_sizes, int n_in,
                              void* d_out, int out_size, void* d_ws, size_t ws_size,
                              hipStream_t stream) {
    const float* x     = (const float*)d_in[0];
    const int*   ei    = (const int*)d_in[1];
    const int*   batch = (const int*)d_in[2];
    const float* ew    = (const float*)d_in[3];
    const float* W1 = (const float*)d_in[4];  const float* b1 = (const float*)d_in[5];
    const float* W2 = (const float*)d_in[6];  const float* b2 = (const float*)d_in[7];
    const float* W3 = (const float*)d_in[8];  const float* b3 = (const float*)d_in[9];
    const float* g1 = (const float*)d_in[10]; const float* bt1 = (const float*)d_in[11];
    const float* g2 = (const float*)d_in[12]; const float* bt2 = (const float*)d_in[13];
    const float* g3 = (const float*)d_in[14]; const float* bt3 = (const float*)d_in[15];
    const float* fcW = (const float*)d_in[16]; const float* fcb = (const float*)d_in[17];

    const int N  = in_sizes[2];               // 100000 nodes (batch vector length)
    const int NE = in_sizes[1] / 2;           // 1600000 edges
    const int NG = (out_size - N * 64) / 66;  // 2048 graphs

    const int* row = ei;
    const int* col = ei + NE;

    // workspace layout
    float* dinv = (float*)d_ws;                       // N floats (deg -> dinv in place)
    float* bufA = dinv + ((N + 63) & ~63);            // N*64
    float* bufB = bufA + (size_t)N * 64;              // N*64
    float* sums = bufB + (size_t)N * 64;              // 128 floats

    // output layout
    float* logits    = (float*)d_out;                 // NG*2
    float* node_embs = logits + (size_t)NG * 2;       // N*64
    float* gemb      = node_embs + (size_t)N * 64;    // NG*64

    const int T = 256;
    const int gN     = (N + T - 1) / T;
    const int gNE    = (NE + T - 1) / T;
    const int gNF    = (N * 64 + T - 1) / T;
    const int gEdgeW = (NE * 32 + T - 1) / T;   // wave-per-edge
    const int gPool  = (N * 32 + T - 1) / T;
    const int gGemm  = (N + 31) / 32;

    // ---- degree -> dinv (edge structure is layer-invariant) ----
    deg_init_kernel<<<gN, T, 0, stream>>>(dinv, N);
    deg_acc_kernel<<<gNE, T, 0, stream>>>(col, ew, dinv, NE);
    deg_inv_kernel<<<gN, T, 0, stream>>>(dinv, N);

    // ---- layer 1: x[N x 14] -> bufA ----
    gemm_wmma_kernel<14, 16><<<gGemm, T, 0, stream>>>(x, W1, bufB, N);
    selfloop_init_kernel<<<gNF, T, 0, stream>>>(bufB, dinv, b1, bufA, N);
    edge_scatter_kernel<<<gEdgeW, T, 0, stream>>>(row, col, ew, dinv, bufB, bufA, NE);
    zero_kernel<<<1, 128, 0, stream>>>(sums, 128);
    bn_stats_kernel<<<512, T, 0, stream>>>(bufA, sums, N);
    bn_relu_kernel<<<gNF, T, 0, stream>>>(bufA, sums, g1, bt1, bufA, N);

    // ---- layer 2 ----
    gemm_wmma_kernel<64, 64><<<gGemm, T, 0, stream>>>(bufA, W2, bufB, N);
    selfloop_init_kernel<<<gNF, T, 0, stream>>>(bufB, dinv, b2, bufA, N);
    edge_scatter_kernel<<<gEdgeW, T, 0, stream>>>(row, col, ew, dinv, bufB, bufA, NE);
    zero_kernel<<<1, 128, 0, stream>>>(sums, 128);
    bn_stats_kernel<<<512, T, 0, stream>>>(bufA, sums, N);
    bn_relu_kernel<<<gNF, T, 0, stream>>>(bufA, sums, g2, bt2, bufA, N);

    // ---- layer 3 (BN output goes straight to node_embs in d_out) ----
    gemm_wmma_kernel<64, 64><<<gGemm, T, 0, stream>>>(bufA, W3, bufB, N);
    selfloop_init_kernel<<<gNF, T, 0, stream>>>(bufB, dinv, b3, bufA, N);
    edge_scatter_kernel<<<gEdgeW, T, 0, stream>>>(row, col, ew, dinv, bufB, bufA, NE);
    zero_kernel<<<1, 128, 0, stream>>>(sums, 128);
    bn_stats_kernel<<<512, T, 0, stream>>>(bufA, sums, N);
    bn_relu_kernel<<<gNF, T, 0, stream>>>(bufA, sums, g3, bt3, node_embs, N);

    // ---- pooling + FC ----
    zero_kernel<<<(NG * 64 + T - 1) / T, T, 0, stream>>>(gemb, NG * 64);
    pool_kernel<<<gPool, T, 0, stream>>>(node_embs, batch, gemb, N);
    fc_kernel<<<(NG * 2 + T - 1) / T, T, 0, stream>>>(gemb, fcW, fcb, logits, NG);
}